// Attention_25786983645717
// MI455X (gfx1250) — compile-verified
//
#include <hip/hip_runtime.h>

// Cross-attention (fp32) for MI455X / gfx1250, wave32, WMMA f32 16x16x4.
// B=32, T_DEC=1024, T_SRC=2048, D=1024.
// Pass 1: per-row softmax stats (max, sum-exp) via WMMA score tiles.
// Pass 2: recompute scores, write normalized attention weights.
// Pass 3: context = weights x encoder GEMM via WMMA.
// Each wave handles a 16x32 tile pair: A fragments are reused across two
// WMMAs, and two independent accumulators per tile break the C->D chain.

typedef __attribute__((ext_vector_type(2))) float v2f;
typedef __attribute__((ext_vector_type(8))) float v8f;

#define BATCH  32
#define TDEC   1024
#define TSRC   2048
#define DDIM   1024
#define NEG_BIG (-1.0e30f)

__device__ __forceinline__ v8f wmma_f32(v2f a, v2f b, v8f c) {
    // 8 args: (neg_a, A, neg_b, B, c_mod, C, reuse_a, reuse_b)
    return __builtin_amdgcn_wmma_f32_16x16x4_f32(
        false, a, false, b, (short)0, c, false, false);
}

__device__ __forceinline__ v2f ldv2(const float* p) { return *(const v2f*)p; }

__device__ __forceinline__ float half_max16(float v) {
    #pragma unroll
    for (int off = 1; off < 16; off <<= 1)
        v = fmaxf(v, __shfl_xor(v, off, 32));
    return v;
}

__device__ __forceinline__ float half_sum16(float v) {
    #pragma unroll
    for (int off = 1; off < 16; off <<= 1)
        v += __shfl_xor(v, off, 32);
    return v;
}

// Score tile pair: sA = Q*K^T for src cols [s0,s0+16), sB for [s0+16,s0+32).
// Q fragment (A operand) is loaded once and feeds both tiles; two
// accumulators per tile remove the WMMA RAW-on-C serialization.
__device__ __forceinline__ void score_pair(const float* __restrict__ qp,
                                           const float* __restrict__ kp0,
                                           const float* __restrict__ kp1,
                                           v8f& sA, v8f& sB) {
    v8f a0 = {}, a1 = {}, b0 = {}, b1 = {};
    #pragma unroll 4
    for (int d = 0; d < DDIM; d += 8) {
        v2f qa = ldv2(qp + d);
        v2f qb = ldv2(qp + d + 4);
        a0 = wmma_f32(qa, ldv2(kp0 + d), a0);
        b0 = wmma_f32(qa, ldv2(kp1 + d), b0);
        a1 = wmma_f32(qb, ldv2(kp0 + d + 4), a1);
        b1 = wmma_f32(qb, ldv2(kp1 + d + 4), b1);
    }
    sA = a0 + a1;
    sB = b0 + b1;
}

// ---------------------------------------------------------------- pass 1
__global__ __launch_bounds__(128) void attn_stats_kernel(
    const float* __restrict__ dec, const float* __restrict__ enc,
    const unsigned char* __restrict__ mask, float* __restrict__ stats) {
    const int lane = threadIdx.x & 31;
    const int wave = threadIdx.x >> 5;
    const int tile = blockIdx.x * 4 + wave;          // one 16-dec-row tile per wave
    const int b    = tile >> 6;                      // / (TDEC/16)
    const int t0   = (tile & 63) << 4;
    const int half = lane >> 4;
    const int l16  = lane & 15;
    const int kb   = half * 2;

    const float* qp = dec + (size_t)(b * TDEC + t0 + l16) * DDIM + kb;
    const unsigned char* mrow = mask + (size_t)b * TSRC;

    float mrun[8], lrun[8];
    #pragma unroll
    for (int r = 0; r < 8; ++r) { mrun[r] = NEG_BIG; lrun[r] = 0.f; }

    for (int s0 = 0; s0 < TSRC; s0 += 32) {
        const float* kp0 = enc + (size_t)(b * TSRC + s0 + l16) * DDIM + kb;
        const float* kp1 = kp0 + (size_t)16 * DDIM;
        // hoisted: mask bytes + prefetch of next tile pair overlap the WMMAs
        const bool valid0 = mrow[s0 + l16] != 0;
        const bool valid1 = mrow[s0 + 16 + l16] != 0;
        if (s0 + 32 < TSRC) {
            __builtin_prefetch(kp0 + (size_t)32 * DDIM, 0, 3);
            __builtin_prefetch(kp0 + (size_t)48 * DDIM, 0, 3);
        }
        v8f sA, sB;
        score_pair(qp, kp0, kp1, sA, sB);

        #pragma unroll
        for (int r = 0; r < 8; ++r) {
            float x0 = valid0 ? sA[r] : NEG_BIG;
            float x1 = valid1 ? sB[r] : NEG_BIG;
            float tmax = half_max16(fmaxf(x0, x1));
            float mnew = fmaxf(mrun[r], tmax);
            float esum = half_sum16(__expf(x0 - mnew) + __expf(x1 - mnew));
            lrun[r] = lrun[r] * __expf(mrun[r] - mnew) + esum;
            mrun[r] = mnew;
        }
    }
    if (l16 == 0) {
        #pragma unroll
        for (int r = 0; r < 8; ++r) {
            const int t = t0 + r + half * 8;
            stats[(size_t)(b * TDEC + t) * 2 + 0] = mrun[r];
            stats[(size_t)(b * TDEC + t) * 2 + 1] = lrun[r];
        }
    }
}

// ---------------------------------------------------------------- pass 2
__global__ __launch_bounds__(128) void attn_weights_kernel(
    const float* __restrict__ dec, const float* __restrict__ enc,
    const unsigned char* __restrict__ mask, const float* __restrict__ stats,
    float* __restrict__ wout) {
    const int lane = threadIdx.x & 31;
    const int wave = threadIdx.x >> 5;
    const int tile = blockIdx.x * 4 + wave;
    const int b    = tile >> 6;
    const int t0   = (tile & 63) << 4;
    const int half = lane >> 4;
    const int l16  = lane & 15;
    const int kb   = half * 2;

    const float* qp = dec + (size_t)(b * TDEC + t0 + l16) * DDIM + kb;
    const unsigned char* mrow = mask + (size_t)b * TSRC;

    float mr[8], linv[8];
    #pragma unroll
    for (int r = 0; r < 8; ++r) {
        const int t = t0 + r + half * 8;
        mr[r]   = stats[(size_t)(b * TDEC + t) * 2 + 0];
        linv[r] = 1.0f / stats[(size_t)(b * TDEC + t) * 2 + 1];
    }

    for (int s0 = 0; s0 < TSRC; s0 += 32) {
        const float* kp0 = enc + (size_t)(b * TSRC + s0 + l16) * DDIM + kb;
        const float* kp1 = kp0 + (size_t)16 * DDIM;
        const bool valid0 = mrow[s0 + l16] != 0;
        const bool valid1 = mrow[s0 + 16 + l16] != 0;
        if (s0 + 32 < TSRC) {
            __builtin_prefetch(kp0 + (size_t)32 * DDIM, 0, 3);
            __builtin_prefetch(kp0 + (size_t)48 * DDIM, 0, 3);
        }
        v8f sA, sB;
        score_pair(qp, kp0, kp1, sA, sB);

        float* wp = wout + (size_t)(b * TDEC + t0 + half * 8) * TSRC + s0 + l16;
        #pragma unroll
        for (int r = 0; r < 8; ++r) {
            const float w0 = valid0 ? __expf(sA[r] - mr[r]) * linv[r] : 0.0f;
            const float w1 = valid1 ? __expf(sB[r] - mr[r]) * linv[r] : 0.0f;
            wp[(size_t)r * TSRC]      = w0;
            wp[(size_t)r * TSRC + 16] = w1;
        }
    }
}

// ---------------------------------------------------------------- pass 3
// Each wave: 16 dec rows x 32 d cols. W (A operand) fragment reused for the
// two d tiles; the two tiles' B loads cover one full 128B line per (s,half).
__global__ __launch_bounds__(256) void attn_context_kernel(
    const float* __restrict__ w, const float* __restrict__ enc,
    float* __restrict__ ctx) {
    const int lane = threadIdx.x & 31;
    const int wave = threadIdx.x >> 5;
    const int tile = blockIdx.x * 8 + wave;
    const int dt   = DDIM / 32;                      // 32
    const int tt   = TDEC / 16;                      // 64
    const int b    = tile / (dt * tt);
    const int rem  = tile % (dt * tt);
    const int t0   = (rem / dt) * 16;
    const int d0   = (rem % dt) * 32;
    const int half = lane >> 4;
    const int l16  = lane & 15;
    const int kb   = half * 2;

    const float* ap  = w   + (size_t)(b * TDEC + t0 + l16) * TSRC + kb;
    const float* bp0 = enc + (size_t)b * TSRC * DDIM + d0 + l16;
    const float* bp1 = bp0 + 16;

    v8f c0a = {}, c0b = {}, c1a = {}, c1b = {};
    #pragma unroll 4
    for (int s = 0; s < TSRC; s += 8) {
        v2f a0 = ldv2(ap + s);
        v2f a1 = ldv2(ap + s + 4);
        v2f b00, b10, b01, b11;
        b00.x = bp0[(size_t)(s + kb + 0) * DDIM];
        b00.y = bp0[(size_t)(s + kb + 1) * DDIM];
        b10.x = bp1[(size_t)(s + kb + 0) * DDIM];
        b10.y = bp1[(size_t)(s + kb + 1) * DDIM];
        b01.x = bp0[(size_t)(s + 4 + kb + 0) * DDIM];
        b01.y = bp0[(size_t)(s + 4 + kb + 1) * DDIM];
        b11.x = bp1[(size_t)(s + 4 + kb + 0) * DDIM];
        b11.y = bp1[(size_t)(s + 4 + kb + 1) * DDIM];
        c0a = wmma_f32(a0, b00, c0a);
        c1a = wmma_f32(a0, b10, c1a);
        c0b = wmma_f32(a1, b01, c0b);
        c1b = wmma_f32(a1, b11, c1b);
    }
    const v8f c0 = c0a + c0b;
    const v8f c1 = c1a + c1b;
    float* cp = ctx + (size_t)(b * TDEC + t0 + half * 8) * DDIM + d0 + l16;
    #pragma unroll
    for (int r = 0; r < 8; ++r) {
        cp[(size_t)r * DDIM]      = c0[r];
        cp[(size_t)r * DDIM + 16] = c1[r];
    }
}

extern "C" void kernel_launch(void* const* d_in, const int* in_sizes, int n_in,
                              void* d_out, int out_size, void* d_ws, size_t ws_size,
                              hipStream_t stream) {
    const float* dec = (const float*)d_in[0];          // [B, TDEC, D] f32
    const float* enc = (const float*)d_in[1];          // [B, TSRC, D] f32
    const unsigned char* mask = (const unsigned char*)d_in[2]; // [B,1,TSRC] bool
    float* ctx = (float*)d_out;                                   // [B,TDEC,D]
    float* wts = (float*)d_out + (size_t)BATCH * TDEC * DDIM;     // [B,TDEC,TSRC]
    float* stats = (float*)d_ws;                                  // [B,TDEC,2] = 256 KB

    const int tilesA = BATCH * (TDEC / 16);            // 2048 wave-tiles
    attn_stats_kernel<<<tilesA / 4, 128, 0, stream>>>(dec, enc, mask, stats);
    attn_weights_kernel<<<tilesA / 4, 128, 0, stream>>>(dec, enc, mask, stats, wts);

    const int tilesC = BATCH * (TDEC / 16) * (DDIM / 32);  // 65536 wave-tiles
    attn_context_kernel<<<tilesC / 8, 256, 0, stream>>>(wts, enc, ctx);
}